// ZBLRepulsion_25924422598830
// MI455X (gfx1250) — compile-verified
//
#include <hip/hip_runtime.h>
#include <hip/hip_bf16.h>
#include <stdint.h>

// ZBL repulsion on gfx1250 (wave32) with TDM-staged displacement tiles.
// d_in order (from setup_inputs):
//  0 atomic_numbers [N] f32 | 1 displacements [P,3] f32 | 2 idx_i [P] i32 (sorted)
//  3 idx_j [P] i32 | 4 atom_mask [N] f32 | 5 batch_mask [P] f32
//  6 batch_segments (unused) | 7 batch_size (unused)
//  8 a_coefficient [1] | 9 a_exponent [1] | 10 phi_coefficients [4] | 11 phi_exponents [4]
// out: Erep [N] f32.

#define TILE     256          // pairs per block (one per thread)
#define WAVE_DW  96           // dwords of displacement per wave slice: TILE*3/8

typedef unsigned int u32x4 __attribute__((ext_vector_type(4)));
typedef int          i32x8 __attribute__((ext_vector_type(8)));
typedef int          i32x4 __attribute__((ext_vector_type(4)));

__device__ __forceinline__ float nt_loadf(const float* p) {
  return __builtin_nontemporal_load(p);
}
__device__ __forceinline__ int nt_loadi(const int* p) {
  return __builtin_nontemporal_load(p);
}
__device__ __forceinline__ float fast_rcp(float x) {
  return __builtin_amdgcn_rcpf(x);
}

struct ZblParams {
  float ac;                 // |a_coefficient|
  float rs;                 // 1 / sum(|phi_c|)
  float c0, c1, c2, c3;     // |phi_c|
  float e0, e1, e2, e3;     // max(|phi_e|, 1e-10)
};

__device__ __forceinline__ ZblParams load_params(const float* __restrict__ a_coef_p,
                                                 const float* __restrict__ pc,
                                                 const float* __restrict__ pe) {
  ZblParams P;
  P.ac = fabsf(a_coef_p[0]);
  P.c0 = fabsf(pc[0]); P.c1 = fabsf(pc[1]); P.c2 = fabsf(pc[2]); P.c3 = fabsf(pc[3]);
  P.rs = fast_rcp(fmaxf((P.c0 + P.c1) + (P.c2 + P.c3), 1e-10f));
  P.e0 = fmaxf(fabsf(pe[0]), 1e-10f);
  P.e1 = fmaxf(fabsf(pe[1]), 1e-10f);
  P.e2 = fmaxf(fabsf(pe[2]), 1e-10f);
  P.e3 = fmaxf(fabsf(pe[3]), 1e-10f);
  return P;
}

// Per-pair ZBL repulsion (already multiplied by batch mask).
__device__ __forceinline__ float pair_repulsion(float x, float y, float z, float bm,
                                                float2 zi, float2 zj,
                                                const ZblParams& P) {
  float d = sqrtf(fmaf(x, x, fmaf(y, y, z * z)));
  d = fmaxf(d, 1e-10f);

  // smooth_switch(d, 0, 10): t > 0 always since d > 0
  const float t = d * 0.1f;
  float sw;
  if (t >= 1.0f) {
    sw = 1.0f;
  } else {
    const float tc = fminf(fmaxf(t, 1e-9f), 1.0f - 1e-9f);
    const float f1 = __expf(-fast_rcp(tc));
    const float f2 = __expf(-fast_rcp(1.0f - tc));
    sw = f1 * fast_rcp(f1 + f2);
  }

  const float denom = fmaxf(zi.y + zj.y, 1e-6f);
  float aij = fminf(P.ac * fast_rcp(denom), 1e6f);
  float arg = fminf(d * fast_rcp(fmaxf(aij, 1e-10f)), 1e6f);

  const float l0 = -P.e0 * arg, l1 = -P.e1 * arg, l2 = -P.e2 * arg, l3 = -P.e3 * arg;
  const float m  = fmaxf(fmaxf(l0, l1), fmaxf(l2, l3));
  float phi = (P.c0 * __expf(l0 - m) + P.c1 * __expf(l1 - m) +
               P.c2 * __expf(l2 - m) + P.c3 * __expf(l3 - m)) * P.rs * __expf(m);
  phi = fmaxf(phi, 1e-30f);

  const float cp   = fminf(zi.x * zj.x, 1e4f);
  const float base = fminf(0.5f * cp * fast_rcp(d), 1e6f);
  float r = base * phi * fmaxf(sw, 1e-30f);
  r = fminf(fmaxf(r, 0.0f), 1e6f);
  if (!(r == r)) r = 0.0f;
  return r * bm;
}

// Wave32 segmented reduction over sorted segment ids; one atomic per run.
__device__ __forceinline__ void segmented_atomic_add(float rep, int seg, int lane,
                                                     float* __restrict__ acc) {
  float v = rep;
  int   s = seg;
#pragma unroll
  for (int off = 1; off < 32; off <<= 1) {
    const float vv = __shfl_up(v, off, 32);
    const int   ss = __shfl_up(s, off, 32);
    if (lane >= off && ss == s) v += vv;
  }
  const int next_s = __shfl_down(s, 1, 32);
  const bool is_last = (lane == 31) || (next_s != s);
  if (is_last && s >= 0) atomicAdd(acc + s, v);
}

// ---------------------------------------------------------------------------
// Kernel 1: zero accumulator; build packed {safeZ, za} table (za = Z^|a_exp|).
// ---------------------------------------------------------------------------
__global__ void zbl_init_kernel(const float* __restrict__ Z,
                                const float* __restrict__ a_exp_p,
                                float* __restrict__ acc,
                                float2* __restrict__ zza,
                                int n_atoms) {
  int i = blockIdx.x * blockDim.x + threadIdx.x;
  if (i >= n_atoms) return;
  float ae = fabsf(a_exp_p[0]);
  float sz = fmaxf(Z[i], 1e-6f);
  float za = __expf(__logf(sz) * ae);
  if (!(za == za)) za = 1e-6f;
  za = fminf(za, 1e6f);
  acc[i] = 0.0f;
  zza[i] = make_float2(sz, za);
}

// ---------------------------------------------------------------------------
// Kernel 2a: full tiles — TDM stages the AoS displacement tile into LDS.
// Each wave DMAs its contiguous 96-dword slice (1-D tensor descriptor), then
// issues ALL of its global loads (streams + random L2 gathers) while the DMA
// is in flight, and only then waits TENSORcnt + barrier. LDS read-back is
// stride-3 -> bank-conflict-free (gcd(3,64)=1).
// ---------------------------------------------------------------------------
__global__ void zbl_pair_tdm_kernel(const float* __restrict__ disp,
                                    const int*  __restrict__ idx_i,
                                    const int*  __restrict__ idx_j,
                                    const float* __restrict__ bmask,
                                    const float2* __restrict__ zza,
                                    const float* __restrict__ a_coef_p,
                                    const float* __restrict__ phi_c,
                                    const float* __restrict__ phi_e,
                                    float* __restrict__ acc) {
  __shared__ float sdisp[TILE * 3];

  const int t    = threadIdx.x;
  const int lane = t & 31;
  const int wave = t >> 5;

  const unsigned long long tile_dw =
      (unsigned long long)blockIdx.x * (unsigned long long)(TILE * 3);

  // ---- 1) Issue the tensor DMA first (TENSORcnt++).
#if __has_builtin(__builtin_amdgcn_tensor_load_to_lds)
  {
    // Tensor DMA descriptor (ISA 8.3/8.4): 1-D tile of WAVE_DW dwords.
    const unsigned lds_addr =
        (unsigned)(unsigned long long)(uintptr_t)(&sdisp[0]) +
        (unsigned)wave * (WAVE_DW * 4u);
    const unsigned long long ga =
        (unsigned long long)(uintptr_t)disp +
        (tile_dw + (unsigned long long)wave * WAVE_DW) * 4ull;

    u32x4 g0;
    g0[0] = 1u;                                        // count=1, normal mode
    g0[1] = lds_addr;                                  // lds_addr (bytes)
    g0[2] = (unsigned)(ga & 0xFFFFFFFFull);            // global_addr[31:0]
    g0[3] = (unsigned)((ga >> 32) & 0x01FFFFFFu)       // global_addr[56:32]
            | 0x80000000u;                             // type=2 ("image")

    i32x8 g1;
    g1[0] = (int)(2u << 16);                           // data_size = 4 B
    g1[1] = (int)(((unsigned)WAVE_DW) << 16);          // tensor_dim0 = 96
    g1[2] = (int)(1u << 16);                           // tensor_dim1 = 1
    g1[3] = (int)(((unsigned)WAVE_DW) << 16);          // tile_dim0 = 96
    g1[4] = 0;                                         // tile_dim1 = tile_dim2 = 0 (1-D)
    g1[5] = WAVE_DW;                                   // tensor_dim0_stride
    g1[6] = 0;
    g1[7] = 0;

    i32x4 gz = {0, 0, 0, 0};                           // groups 2/3: <=2-D tensor
#if defined(__clang_major__) && __clang_major__ >= 23
    i32x8 gz8 = {0, 0, 0, 0, 0, 0, 0, 0};
    __builtin_amdgcn_tensor_load_to_lds(g0, g1, gz, gz, gz8, 0);
#else
    __builtin_amdgcn_tensor_load_to_lds(g0, g1, gz, gz, 0);
#endif
  }
#else
  {
    // Fallback: coalesced staged loads (3 fully-coalesced dword loads/thread).
    const float* src = disp + tile_dw;
    sdisp[t]           = nt_loadf(src + t);
    sdisp[t + TILE]    = nt_loadf(src + t + TILE);
    sdisp[t + 2*TILE]  = nt_loadf(src + t + 2 * TILE);
  }
#endif

  // ---- 2) Overlap: issue every independent global load while the DMA runs.
  const int p  = blockIdx.x * TILE + t;                // full tiles only
  const float bm = nt_loadf(bmask + p);                // NT stream
  const int seg  = nt_loadi(idx_i + p);                // sorted NT stream
  const int j    = nt_loadi(idx_j + p);                // random NT stream
  const float2 zi = zza[seg];                          // L2-resident gather
  const float2 zj = zza[j];                            // L2-resident gather
  const ZblParams P = load_params(a_coef_p, phi_c, phi_e);  // uniform scalars

  // ---- 3) Now drain the DMA and make the tile visible to all waves.
#if __has_builtin(__builtin_amdgcn_tensor_load_to_lds)
  __builtin_amdgcn_s_wait_tensorcnt(0);
#endif
  __syncthreads();

  const float sh = 1.0f - bm;
  const float x = sdisp[3 * t + 0] + sh;               // stride-3: conflict-free
  const float y = sdisp[3 * t + 1] + sh;
  const float z = sdisp[3 * t + 2] + sh;

  const float rep = pair_repulsion(x, y, z, bm, zi, zj, P);
  segmented_atomic_add(rep, seg, lane, acc);
}

// ---------------------------------------------------------------------------
// Kernel 2b: tail pairs (direct loads) — covers n_pairs % TILE remainder.
// ---------------------------------------------------------------------------
__global__ void zbl_pair_tail_kernel(const float* __restrict__ disp,
                                     const int*  __restrict__ idx_i,
                                     const int*  __restrict__ idx_j,
                                     const float* __restrict__ bmask,
                                     const float2* __restrict__ zza,
                                     const float* __restrict__ a_coef_p,
                                     const float* __restrict__ phi_c,
                                     const float* __restrict__ phi_e,
                                     float* __restrict__ acc,
                                     int base, int n_pairs) {
  const int p    = base + blockIdx.x * blockDim.x + threadIdx.x;
  const int lane = threadIdx.x & 31;

  float rep = 0.0f;
  int   seg = -1;
  if (p < n_pairs) {
    const ZblParams P = load_params(a_coef_p, phi_c, phi_e);
    const float bm = nt_loadf(bmask + p);
    const float sh = 1.0f - bm;
    const float x = nt_loadf(disp + 3 * p + 0) + sh;
    const float y = nt_loadf(disp + 3 * p + 1) + sh;
    const float z = nt_loadf(disp + 3 * p + 2) + sh;
    seg = nt_loadi(idx_i + p);
    const int j = nt_loadi(idx_j + p);
    rep = pair_repulsion(x, y, z, bm, zza[seg], zza[j], P);
  }
  segmented_atomic_add(rep, seg, lane, acc);
}

// ---------------------------------------------------------------------------
// Kernel 3: mask, clip, *0.01 -> output.
// ---------------------------------------------------------------------------
__global__ void zbl_finalize_kernel(const float* __restrict__ acc,
                                    const float* __restrict__ amask,
                                    float* __restrict__ out,
                                    int n_atoms) {
  int i = blockIdx.x * blockDim.x + threadIdx.x;
  if (i >= n_atoms) return;
  float v = acc[i] * amask[i];
  v = fminf(fmaxf(v, 0.0f), 1e6f);
  if (!(v == v)) v = 0.0f;
  __builtin_nontemporal_store(v * 0.01f, out + i);
}

// ---------------------------------------------------------------------------
extern "C" void kernel_launch(void* const* d_in, const int* in_sizes, int n_in,
                              void* d_out, int out_size, void* d_ws, size_t ws_size,
                              hipStream_t stream) {
  const float* Z      = (const float*)d_in[0];
  const float* disp   = (const float*)d_in[1];
  const int*   idx_i  = (const int*)d_in[2];
  const int*   idx_j  = (const int*)d_in[3];
  const float* amask  = (const float*)d_in[4];
  const float* bmask  = (const float*)d_in[5];
  const float* a_coef = (const float*)d_in[8];
  const float* a_exp  = (const float*)d_in[9];
  const float* phi_c  = (const float*)d_in[10];
  const float* phi_e  = (const float*)d_in[11];

  const int n_atoms = in_sizes[0];
  const int n_pairs = in_sizes[2];

  // Workspace: [acc: n_atoms f32][pad to 256B][zza: n_atoms float2]
  float*  acc = (float*)d_ws;
  size_t  off = ((size_t)n_atoms * sizeof(float) + 255u) & ~(size_t)255u;
  float2* zza = (float2*)((char*)d_ws + off);

  const int blk = TILE;
  const int gA  = (n_atoms + blk - 1) / blk;

  const int full_tiles = n_pairs / TILE;           // 25000 for P = 6.4M
  const int tail_base  = full_tiles * TILE;
  const int tail       = n_pairs - tail_base;

  zbl_init_kernel<<<gA, blk, 0, stream>>>(Z, a_exp, acc, zza, n_atoms);
  if (full_tiles > 0) {
    zbl_pair_tdm_kernel<<<full_tiles, blk, 0, stream>>>(
        disp, idx_i, idx_j, bmask, zza, a_coef, phi_c, phi_e, acc);
  }
  if (tail > 0) {
    zbl_pair_tail_kernel<<<(tail + blk - 1) / blk, blk, 0, stream>>>(
        disp, idx_i, idx_j, bmask, zza, a_coef, phi_c, phi_e, acc,
        tail_base, n_pairs);
  }
  zbl_finalize_kernel<<<gA, blk, 0, stream>>>(acc, amask, (float*)d_out, n_atoms);
}